// LightGCN_52080773431354
// MI455X (gfx1250) — compile-verified
//
#include <hip/hip_runtime.h>
#include <stdint.h>

// ---------------------------------------------------------------------------
// LightGCN propagation on MI455X (gfx1250).
// Memory-bound SpMM (AI ~0.27 FLOP/B): no WMMA (random sparsity, no dense
// tiles). Strategy: build CSR once (int atomics), then 3 atomic-free
// row-parallel SpMM passes whose random gathers hit the 192MB L2 (x = 51MB).
// CDNA5-specific: async global->LDS edge streaming (ASYNCcnt) in the CSR
// scatter pass, global prefetch in the SpMM edge loop, wave32 16-lane
// row-groups.
// Workspace needed: ~2*N*D*4 + 2*E*4 + (2N+1)*4  (~130 MB for this problem).
// ---------------------------------------------------------------------------

#ifndef __has_builtin
#define __has_builtin(x) 0
#endif

#if __has_builtin(__builtin_amdgcn_global_load_async_to_lds_b128)
#define HAVE_ASYNC_LDS 1
#else
#define HAVE_ASYNC_LDS 0
#endif
#if __has_builtin(__builtin_amdgcn_s_wait_asynccnt)
#define HAVE_WAIT_ASYNC 1
#else
#define HAVE_WAIT_ASYNC 0
#endif

// The builtin expects pointers to 16-byte int vectors in the global (AS1) and
// LDS (AS3) address spaces: (v4i AS1* src, v4i AS3* dst, imm offset, imm cpol).
typedef int v4i __attribute__((vector_size(16)));
typedef __attribute__((address_space(1))) v4i as1_v4i;
typedef __attribute__((address_space(3))) v4i as3_v4i;

// Stage 16 bytes global -> LDS, asynchronously if the gfx1250 builtin exists.
__device__ __forceinline__ void stage16(void* lds_dst, const void* gsrc) {
#if HAVE_ASYNC_LDS
  __builtin_amdgcn_global_load_async_to_lds_b128(
      (as1_v4i*)const_cast<void*>(gsrc), (as3_v4i*)lds_dst,
      /*offset=*/0, /*cpol=*/0);
#else
  *reinterpret_cast<int4*>(lds_dst) = *reinterpret_cast<const int4*>(gsrc);
#endif
}

template <int N>
__device__ __forceinline__ void async_wait() {
#if HAVE_WAIT_ASYNC
  __builtin_amdgcn_s_wait_asynccnt(N);
#elif HAVE_ASYNC_LDS
  asm volatile("s_wait_asynccnt %0" ::"n"(N) : "memory");
#endif
}

// ---------------------------------------------------------------------------
// 1) zero an int array
__global__ void k_zero_i32(int* __restrict__ p, int n) {
  int i = blockIdx.x * blockDim.x + threadIdx.x;
  if (i < n) p[i] = 0;
}

// 2) per-row edge histogram; counts land at row_ptr[r+1]
__global__ void k_hist(const int* __restrict__ rows, int* __restrict__ row_ptr,
                       int E) {
  for (int i = blockIdx.x * blockDim.x + threadIdx.x; i < E;
       i += gridDim.x * blockDim.x) {
    atomicAdd(&row_ptr[rows[i] + 1], 1);
  }
}

// 3) single-workgroup inclusive scan over n ints (tiles of 1024, LDS scan)
__global__ void k_scan(int* __restrict__ data, int n) {
  __shared__ int buf[1024];
  __shared__ int carry;
  const int tid = threadIdx.x;
  if (tid == 0) carry = 0;
  __syncthreads();
  for (int base = 0; base < n; base += 1024) {
    int i = base + tid;
    int v = (i < n) ? data[i] : 0;
    buf[tid] = v;
    __syncthreads();
    for (int off = 1; off < 1024; off <<= 1) {
      int t = (tid >= off) ? buf[tid - off] : 0;
      __syncthreads();
      buf[tid] += t;
      __syncthreads();
    }
    int outv = buf[tid] + carry;
    if (i < n) data[i] = outv;
    __syncthreads();
    if (tid == 1023) carry = outv;
    __syncthreads();
  }
}

// 4) cursor = row_ptr (per-row write cursors for the scatter pass)
__global__ void k_copy_i32(int* __restrict__ dst, const int* __restrict__ src,
                           int n) {
  int i = blockIdx.x * blockDim.x + threadIdx.x;
  if (i < n) dst[i] = src[i];
}

// 5) scatter edges into CSR order. Edge stream (row/col/val) is staged
//    through LDS with double-buffered async global->LDS B128 copies.
#define SC_TILE 1024  // edges per tile; 256 threads * 4 int/thread per array
__global__ void __launch_bounds__(256) k_scatter(
    const int* __restrict__ rows, const int* __restrict__ cols,
    const float* __restrict__ vals, int* __restrict__ cursor,
    int* __restrict__ scol, float* __restrict__ sval, int E) {
  __shared__ int   sh_row[2][SC_TILE];
  __shared__ int   sh_col[2][SC_TILE];
  __shared__ float sh_val[2][SC_TILE];
  const int tid   = threadIdx.x;
  const int nFull = E / SC_TILE;

  auto stage_tile = [&](int tile, int b) {
    const size_t base = (size_t)tile * SC_TILE + (size_t)tid * 4;
    stage16(&sh_row[b][tid * 4], rows + base);
    stage16(&sh_col[b][tid * 4], cols + base);
    stage16(&sh_val[b][tid * 4], vals + base);
  };

  int t = blockIdx.x;
  if (t < nFull) stage_tile(t, 0);
  int b = 0;
  for (; t < nFull; t += gridDim.x) {
    const int tn = t + gridDim.x;
    if (tn < nFull) {
      stage_tile(tn, b ^ 1);
      async_wait<3>();  // 3 newest (next tile) may remain in flight
    } else {
      async_wait<0>();  // drain: current tile must be resident
    }
    __syncthreads();
#pragma unroll
    for (int k = 0; k < 4; ++k) {
      const int i    = tid * 4 + k;
      const int r    = sh_row[b][i];
      const int slot = atomicAdd(&cursor[r], 1);
      scol[slot] = sh_col[b][i];
      sval[slot] = sh_val[b][i];
    }
    __syncthreads();  // everyone done reading buffer b before it is re-staged
    b ^= 1;
  }
  // tail edges (E % SC_TILE) via plain loads
  for (int i = nFull * SC_TILE + blockIdx.x * blockDim.x + tid; i < E;
       i += gridDim.x * blockDim.x) {
    const int r    = rows[i];
    const int slot = atomicAdd(&cursor[r], 1);
    scol[slot] = cols[i];
    sval[slot] = vals[i];
  }
}

// 6) out = emb; x = emb   (vectorized float4 streams)
__global__ void k_init(const float4* __restrict__ emb, float4* __restrict__ out,
                       float4* __restrict__ x, int n4) {
  int i = blockIdx.x * blockDim.x + threadIdx.x;
  if (i < n4) {
    float4 v = emb[i];
    out[i] = v;
    x[i]   = v;
  }
}

// 7) atomic-free row-parallel SpMM: y[r,:] = sum_e val_e * x[col_e,:]
//    16-lane group per row (2 rows / wave32), float4 per lane = 64 dims.
__global__ void __launch_bounds__(256) k_spmm(
    const int* __restrict__ row_ptr, const int* __restrict__ scol,
    const float* __restrict__ sval, const float* __restrict__ x,
    float* __restrict__ y, int N) {
  const int r = blockIdx.x * 16 + (threadIdx.x >> 4);
  if (r >= N) return;  // whole 16-lane group exits together
  const int lane = threadIdx.x & 15;
  const int s = row_ptr[r];
  const int e = row_ptr[r + 1];
  float4 acc = make_float4(0.f, 0.f, 0.f, 0.f);
  for (int base = s; base < e; base += 16) {
    const int n = (e - base) < 16 ? (e - base) : 16;
    if (base + 16 < e) {  // keep the edge stream ahead in cache
      __builtin_prefetch(&scol[base + 16], 0, 0);
      __builtin_prefetch(&sval[base + 16], 0, 0);
    }
    int   c = 0;
    float v = 0.f;
    if (lane < n) {
      c = scol[base + lane];
      v = sval[base + lane];
    }
    for (int j = 0; j < n; ++j) {
      const int   cc = __shfl(c, j, 16);
      const float vv = __shfl(v, j, 16);
      const float4 xv =
          *reinterpret_cast<const float4*>(x + (size_t)cc * 64 + lane * 4);
      acc.x += vv * xv.x;
      acc.y += vv * xv.y;
      acc.z += vv * xv.z;
      acc.w += vv * xv.w;
    }
  }
  *reinterpret_cast<float4*>(y + (size_t)r * 64 + lane * 4) = acc;
}

// 8) out += c * y
__global__ void k_axpy(float4* __restrict__ out, const float4* __restrict__ y,
                       float c, int n4) {
  int i = blockIdx.x * blockDim.x + threadIdx.x;
  if (i < n4) {
    float4 o = out[i];
    float4 v = y[i];
    o.x += c * v.x;
    o.y += c * v.y;
    o.z += c * v.z;
    o.w += c * v.w;
    out[i] = o;
  }
}

// ---------------------------------------------------------------------------
extern "C" void kernel_launch(void* const* d_in, const int* in_sizes, int n_in,
                              void* d_out, int out_size, void* d_ws,
                              size_t ws_size, hipStream_t stream) {
  (void)n_in;
  (void)out_size;
  const float* emb     = (const float*)d_in[0];   // [N, 64]
  const float* A_vals  = (const float*)d_in[1];   // [E]
  const int*   A_idx   = (const int*)d_in[2];     // [2, E]: row then col
  const int    N       = in_sizes[0] / 64;
  const int    E       = in_sizes[1];
  const int*   rows    = A_idx;
  const int*   cols    = A_idx + (size_t)E;

  // workspace layout
  auto al = [](size_t x) { return (x + 255) & ~(size_t)255; };
  char*  w       = (char*)d_ws;
  size_t off     = 0;
  int*   row_ptr = (int*)(w + off);   off = al(off + sizeof(int) * (size_t)(N + 1));
  int*   cursor  = (int*)(w + off);   off = al(off + sizeof(int) * (size_t)N);
  int*   scol    = (int*)(w + off);   off = al(off + sizeof(int) * (size_t)E);
  float* sval    = (float*)(w + off); off = al(off + sizeof(float) * (size_t)E);
  float* xa      = (float*)(w + off); off = al(off + sizeof(float) * (size_t)N * 64);
  float* xb      = (float*)(w + off); off = al(off + sizeof(float) * (size_t)N * 64);
  if (off > ws_size) return;  // insufficient workspace

  const int n4      = N * 16;  // N*64/4 float4 elements
  const int TPB     = 256;
  const int blkN1   = (N + 1 + TPB - 1) / TPB;
  const int blkN    = (N + TPB - 1) / TPB;
  const int blkVec  = (n4 + TPB - 1) / TPB;
  const int blkRows = (N + 15) / 16;

  // ---- build CSR once (reused by all 3 layers) ----
  k_zero_i32<<<blkN1, TPB, 0, stream>>>(row_ptr, N + 1);
  k_hist<<<1024, TPB, 0, stream>>>(rows, row_ptr, E);
  k_scan<<<1, 1024, 0, stream>>>(row_ptr, N + 1);
  k_copy_i32<<<blkN, TPB, 0, stream>>>(cursor, row_ptr, N);
  k_scatter<<<512, TPB, 0, stream>>>(rows, cols, A_vals, cursor, scol, sval, E);

  // ---- out = x0 = emb ----
  float* out = (float*)d_out;
  k_init<<<blkVec, TPB, 0, stream>>>((const float4*)emb, (float4*)out,
                                     (float4*)xa, n4);

  // ---- 3 propagation layers: x_{k} = A x_{k-1}; out += x_k/(k+1) ----
  float* xc = xa;
  float* xn = xb;
  for (int k = 1; k <= 3; ++k) {
    k_spmm<<<blkRows, TPB, 0, stream>>>(row_ptr, scol, sval, xc, xn, N);
    k_axpy<<<blkVec, TPB, 0, stream>>>((float4*)out, (const float4*)xn,
                                       1.0f / (float)(k + 1), n4);
    float* t = xc; xc = xn; xn = t;
  }
}